// Gemma3nAudioCumulativeGroupNorm_2783138808552
// MI455X (gfx1250) — compile-verified
//
#include <hip/hip_runtime.h>
#include <hip/hip_bf16.h>
#include <stdint.h>

// Problem shape (fixed by the reference): x[B=8, T=4096, C=1536] fp32.
#define B_   8
#define T_   4096
#define C_   1536
#define ROWS (B_ * T_)          // 32768 rows of length C
#define CHUNK 128               // columns staged per async chunk
#define NCHUNK (C_ / CHUNK)     // 12 chunks per row-tile
#define PSTR 132                // padded LDS row stride (floats): 132 % 64 == 4 -> no bank clash
#define WAVES_PER_BLOCK 2

typedef float v2f __attribute__((ext_vector_type(2)));
typedef float v4f __attribute__((ext_vector_type(4)));
typedef float v8f __attribute__((ext_vector_type(8)));

// ---- CDNA5 async global->LDS copy (ASYNCcnt path), per ISA 15.18.3 op 98 ----
// VDST = VGPR with LDS byte address (low 32 bits of the generic shared pointer
// are the LDS offset per the aperture encoding), VADDR = 64-bit global address.
__device__ __forceinline__ void async_ld_b128(const float* g, const float* l) {
  uint32_t loff = (uint32_t)(uintptr_t)l;
  uint64_t ga   = (uint64_t)(uintptr_t)g;
  asm volatile("global_load_async_to_lds_b128 %0, %1, off"
               :: "v"(loff), "v"(ga) : "memory");
}

// =====================================================================
// Pass 1: per-row s[t] = sum_c x, q[t] = sum_c x^2.
// One wave per 16-row tile; double-buffered async staging of 16x128 chunks.
// Cross-lane K=4 reduction done with V_WMMA_F32_16X16X4_F32 (B = ones).
// =====================================================================
__global__ __launch_bounds__(32 * WAVES_PER_BLOCK)
void k_rowstats(const float* __restrict__ x,
                float* __restrict__ sums,
                float* __restrict__ ssqs) {
  __shared__ __align__(16) float lds[WAVES_PER_BLOCK][2][16 * PSTR];

  const int wave = threadIdx.x >> 5;
  const int lane = threadIdx.x & 31;
  const int tile = blockIdx.x * WAVES_PER_BLOCK + wave;
  const int row0 = tile * 16;

  // Prime chunk 0: 16 async b128 ops (one per row; 32 lanes x float4 = 128 cols).
  {
    float* buf = &lds[wave][0][0];
#pragma unroll
    for (int i = 0; i < 16; ++i)
      async_ld_b128(x + (size_t)(row0 + i) * C_ + 4 * lane,
                    buf + i * PSTR + 4 * lane);
  }

  // WMMA A-matrix (16x4 f32) layout: lanes 0-15 hold {K=0,K=1} for M=lane,
  // lanes 16-31 hold {K=2,K=3} for M=lane-16. Any column->K partition works
  // for a sum, so sub-block k of each chunk (cols [32k,32k+32)) feeds slot k.
  const int r  = lane & 15;
  const int kb = (lane >> 4) * 2;            // 0 for low half-wave, 2 for high
  v2f accS = {0.f, 0.f}, accQ = {0.f, 0.f};

  for (int j = 0; j < NCHUNK; ++j) {
    if (j + 1 < NCHUNK) {
      float* buf = &lds[wave][(j + 1) & 1][0];
#pragma unroll
      for (int i = 0; i < 16; ++i)
        async_ld_b128(x + (size_t)(row0 + i) * C_ + (j + 1) * CHUNK + 4 * lane,
                      buf + i * PSTR + 4 * lane);
      // 16 newer ops may remain outstanding; chunk j (issued earlier, in-order)
      // is complete once <=16 remain.
      asm volatile("s_wait_asynccnt 16" ::: "memory");
    } else {
      asm volatile("s_wait_asynccnt 0" ::: "memory");
    }

    const float* buf = &lds[wave][j & 1][0];
#pragma unroll
    for (int i = 0; i < 8; ++i) {
      v4f a = *(const v4f*)(buf + r * PSTR + (kb + 0) * 32 + 4 * i);
      v4f b = *(const v4f*)(buf + r * PSTR + (kb + 1) * 32 + 4 * i);
      accS.x += a.x + a.y + a.z + a.w;
      accQ.x += a.x * a.x + a.y * a.y + a.z * a.z + a.w * a.w;
      accS.y += b.x + b.y + b.z + b.w;
      accQ.y += b.x * b.x + b.y * b.y + b.z * b.z + b.w * b.w;
    }
  }

  // D = A x ones(4x16): D[m][n] = s[m] for all n. Uniform flow -> EXEC all-1s.
  v2f ones = {1.f, 1.f};
  v8f zero = {};
  v8f ds = __builtin_amdgcn_wmma_f32_16x16x4_f32(false, accS, false, ones,
                                                 (short)0, zero, false, false);
  v8f dq = __builtin_amdgcn_wmma_f32_16x16x4_f32(false, accQ, false, ones,
                                                 (short)0, zero, false, false);

  // C/D layout: VGPR p holds M=p (lanes 0-15, N=lane) and M=p+8 (lanes 16-31).
  // Column N=0 lives in lanes 0 and 16 -> they own s[0..7] / s[8..15].
  if ((lane & 15) == 0) {
    const int off = (lane >> 4) * 8;
#pragma unroll
    for (int p = 0; p < 8; ++p) {
      sums[row0 + off + p] = ds[p];
      ssqs[row0 + off + p] = dq[p];
    }
  }
}

// =====================================================================
// Pass 2: per batch, scan s -> cum_sum -> m[t]; then d[t] = q - 2ms + C m^2,
// scan d -> cum_sq -> scale[t] = rsqrt(cum_sq/cnt + eps). 256 thr x 16 elems.
// =====================================================================
__global__ __launch_bounds__(256)
void k_scan(const float* __restrict__ sums, const float* __restrict__ ssqs,
            float* __restrict__ meanB, float* __restrict__ scaleB) {
  __shared__ float sh[256];
  const int b = blockIdx.x, tid = threadIdx.x;
  const int base = tid * 16;
  const float* s = sums + (size_t)b * T_;
  const float* q = ssqs + (size_t)b * T_;

  float ls[16], lp[16];
  float run = 0.f;
#pragma unroll
  for (int i = 0; i < 16; ++i) { ls[i] = s[base + i]; run += ls[i]; lp[i] = run; }

  sh[tid] = run;
  __syncthreads();
  for (int d = 1; d < 256; d <<= 1) {
    float t = (tid >= d) ? sh[tid - d] : 0.f;
    __syncthreads();
    sh[tid] += t;
    __syncthreads();
  }
  const float excl = sh[tid] - run;

  float mloc[16], dloc[16];
  float run2 = 0.f;
#pragma unroll
  for (int i = 0; i < 16; ++i) {
    const int t = base + i;
    const float cnt = (float)(t + 1) * (float)C_;
    const float m = (excl + lp[i]) / cnt;
    mloc[i] = m;
    const float d = q[t] - 2.f * m * ls[i] + (float)C_ * m * m;
    run2 += d;
    dloc[i] = run2;
  }

  __syncthreads();
  sh[tid] = run2;
  __syncthreads();
  for (int d = 1; d < 256; d <<= 1) {
    float t = (tid >= d) ? sh[tid - d] : 0.f;
    __syncthreads();
    sh[tid] += t;
    __syncthreads();
  }
  const float excl2 = sh[tid] - run2;

#pragma unroll
  for (int i = 0; i < 16; ++i) {
    const int t = base + i;
    const float cnt = (float)(t + 1) * (float)C_;
    const float var = (excl2 + dloc[i]) / cnt;
    meanB[(size_t)b * T_ + t]  = mloc[i];
    scaleB[(size_t)b * T_ + t] = __frsqrt_rn(var + 1e-3f);
  }
}

// =====================================================================
// Pass 3: out = (x - mean[row]) * scale[row] * weight[c]. One block per row.
// x re-read hits L2 (seeded by pass 1); stores are non-temporal so the
// 192 MB output stream doesn't evict x.
// =====================================================================
__global__ __launch_bounds__(128)
void k_norm(const float* __restrict__ x, const float* __restrict__ w,
            const float* __restrict__ meanB, const float* __restrict__ scaleB,
            float* __restrict__ out) {
  const int row = blockIdx.x;
  const float m  = meanB[row];
  const float sc = scaleB[row];
  const size_t base = (size_t)row * C_;
  const v4f* xp = (const v4f*)(x + base);
  const v4f* wp = (const v4f*)w;
  v4f* op = (v4f*)(out + base);
#pragma unroll
  for (int k = 0; k < 3; ++k) {                 // 3 * 128 = 384 float4 = 1536
    const int i = threadIdx.x + 128 * k;
    v4f xv = xp[i];
    v4f wv = wp[i];
    v4f o = (xv - m) * sc * wv;
    __builtin_nontemporal_store(o, op + i);
  }
}

extern "C" void kernel_launch(void* const* d_in, const int* in_sizes, int n_in,
                              void* d_out, int out_size, void* d_ws, size_t ws_size,
                              hipStream_t stream) {
  (void)in_sizes; (void)n_in; (void)out_size; (void)ws_size;
  const float* x = (const float*)d_in[0];
  const float* w = (const float*)d_in[1];
  float* out = (float*)d_out;

  float* sums   = (float*)d_ws;        // [ROWS]
  float* ssqs   = sums   + ROWS;       // [ROWS]
  float* meanB  = ssqs   + ROWS;       // [ROWS]
  float* scaleB = meanB  + ROWS;       // [ROWS]  (512 KB total workspace)

  k_rowstats<<<ROWS / 16 / WAVES_PER_BLOCK, 32 * WAVES_PER_BLOCK, 0, stream>>>(x, sums, ssqs);
  k_scan<<<B_, 256, 0, stream>>>(sums, ssqs, meanB, scaleB);
  k_norm<<<ROWS, 128, 0, stream>>>(x, w, meanB, scaleB, out);
}